// RSNN3_2551210574103
// MI455X (gfx1250) — compile-verified
//
#include <hip/hip_runtime.h>

// Problem constants (match reference)
#define T_STEPS 1000
#define BATCH   256
#define NIN     3
#define NHID    200
#define NWAVE   7           // 7 waves/WG -> 2 waves/SIMD; 2 N-tiles per wave, both streamed from LDS
#define NTILES  (NWAVE*2)   // 14 tiles (13 used, 1 dead)
#define BLOCK   (NWAVE*32)  // 224 threads
#define KC      50          // 200 / 4 K-chunks for V_WMMA_F32_16X16X4_F32
#define KP      25          // K-chunk pairs (one ds_load_b128 feeds two chunks)
#define LSTRIDE 100         // per-lane floats in a fragment stream (400B, 16B aligned)
#define BETA    0.85f
#define THRESH  1.0f

typedef float v2f __attribute__((ext_vector_type(2)));
typedef float v4f __attribute__((ext_vector_type(4)));
typedef float v8f __attribute__((ext_vector_type(8)));

// LDS layout (floats)
#define WRECL_FLOATS (NTILES * 32 * LSTRIDE)  // 44800: B-frags [nt][lane][kc*2+j]
#define SPKF_FLOATS  (2 * 32 * LSTRIDE)       //  6400: A-frags, double buffered
#define XBUF_FLOATS  (2 * 16 * 4)             //   128
#define ACC_FLOATS   (2 * 16)                 //    32
#define LDS_FLOATS   (WRECL_FLOATS + SPKF_FLOATS + XBUF_FLOATS + ACC_FLOATS)
#define LDS_BYTES    (LDS_FLOATS * 4)         // 205,440 B (< 320 KB/WGP)

__global__ __launch_bounds__(BLOCK, 1)
void rsnn_persistent_kernel(const float* __restrict__ x,     // [T,B,3]
                            const float* __restrict__ W1,    // [200,3]
                            const float* __restrict__ b1,    // [200]
                            const float* __restrict__ Wrec,  // [200,200]
                            const float* __restrict__ brec,  // [200]
                            const float* __restrict__ W2,    // [1,200]
                            const float* __restrict__ b2,    // [1]
                            float* __restrict__ out)         // [T*B] cur2 ++ [T*B*200] spk
{
    extern __shared__ float lds[];
    float* wrecl = lds;                     // B-frags of all 14 tiles
    float* spkf  = lds + WRECL_FLOATS;      // spike A-frags (2 bufs)
    float* xbuf  = spkf + SPKF_FLOATS;      // x_t rows (2 bufs)
    float* acc   = xbuf + XBUF_FLOATS;      // cur2 partial sums (2 bufs)

    const int tid   = threadIdx.x;
    const int wave  = tid >> 5;      // 7 waves; tiles nt = wave*2 + s, s in {0,1}
    const int lane  = tid & 31;
    const int h     = lane >> 4;     // wave half
    const int nlo   = lane & 15;     // column within a 16-wide N-tile
    const int bbase = blockIdx.x * 16;

    float* out_cur2 = out;                                   // [T,B]
    float* out_spk  = out + (size_t)T_STEPS * BATCH;         // [T,B,200]

    // ---------------- one-time init ----------------
    for (int i = tid; i < SPKF_FLOATS; i += BLOCK) spkf[i] = 0.0f;   // spk0 = 0
    if (tid < 32) acc[tid] = 0.0f;
    if (tid < 16) {  // stage x_0 into buffer 0
        const float* xr = x + ((size_t)bbase + tid) * NIN;
        xbuf[tid * 4 + 0] = xr[0];
        xbuf[tid * 4 + 1] = xr[1];
        xbuf[tid * 4 + 2] = xr[2];
    }
    // LDS B-fragments: B[k][n] = Wrec[n][k], zero-padded for n >= 200.
    // f32 16x16x4: lane -> N (=ln&15), element j with half (ln>>4) -> K = j + 2*(ln>>4)
    for (int idx = tid; idx < WRECL_FLOATS; idx += BLOCK) {
        int pos = idx % LSTRIDE;                  // kc*2 + j
        int ln  = (idx / LSTRIDE) & 31;
        int nt  = idx / (32 * LSTRIDE);           // 0..13
        int j   = pos & 1;
        int kc  = pos >> 1;
        int n   = nt * 16 + (ln & 15);
        int k   = kc * 4 + ((ln >> 4) << 1) + j;
        wrecl[idx] = (n < NHID) ? Wrec[n * NHID + k] : 0.0f;
    }

    // per-tile constants
    int   ncol[2];  bool nv[2];
    float w10[2], w11[2], w12[2], bsum[2], w2n[2];
    #pragma unroll
    for (int s = 0; s < 2; ++s) {
        int n   = (wave * 2 + s) * 16 + nlo;   // 0..223
        bool v  = (n < NHID);
        ncol[s] = n; nv[s] = v;
        w10[s]  = v ? W1[n * 3 + 0] : 0.0f;
        w11[s]  = v ? W1[n * 3 + 1] : 0.0f;
        w12[s]  = v ? W1[n * 3 + 2] : 0.0f;
        bsum[s] = v ? (b1[n] + brec[n]) : 0.0f;
        w2n[s]  = v ? W2[n] : 0.0f;
    }
    const float b2v = b2[0];

    float mem[2][8];
    #pragma unroll
    for (int s = 0; s < 2; ++s)
        #pragma unroll
        for (int i = 0; i < 8; ++i) mem[s][i] = 0.0f;

    const v8f vzero = {0.f,0.f,0.f,0.f,0.f,0.f,0.f,0.f};

    const float* bptr0 = wrecl + (((wave * 2 + 0) * 32) + lane) * LSTRIDE;
    const float* bptr1 = wrecl + (((wave * 2 + 1) * 32) + lane) * LSTRIDE;

    __syncthreads();

    // ---------------- time loop (sequential recurrence) ----------------
    #pragma unroll 1
    for (int t = 0; t < T_STEPS; ++t) {
        const int rb = t & 1;        // spike buffer entering step t
        const int wb = rb ^ 1;       // spike buffer produced by step t

        // cur2 readout of step t-1 (wave 0) — off the critical recurrence path
        if (wave == 0 && lane < 16 && t > 0) {
            float v = acc[wb * 16 + lane];            // wb == (t-1)&1
            out_cur2[(size_t)(t - 1) * BATCH + bbase + lane] = v + b2v;
            acc[wb * 16 + lane] = 0.0f;
        }
        // prefetch x_{t+1} (wave 1) into the other x buffer
        if (wave == 1 && lane < 16 && (t + 1) < T_STEPS) {
            const float* xr = x + (((size_t)(t + 1)) * BATCH + bbase + lane) * NIN;
            xbuf[(wb * 16 + lane) * 4 + 0] = xr[0];
            xbuf[(wb * 16 + lane) * 4 + 1] = xr[1];
            xbuf[(wb * 16 + lane) * 4 + 2] = xr[2];
        }

        // ---- rec tiles = spk @ Wrec^T : per kp: 3 ds_load_b128 + 4 WMMA,
        //      two independent accumulator chains ----
        const float* aptr = spkf + ((rb * 32) + lane) * LSTRIDE;

        v8f c0 = vzero, c1 = vzero;
        #pragma unroll 5
        for (int kp = 0; kp < KP; ++kp) {
            v4f a  = *(const v4f*)(aptr  + kp * 4);
            v4f b0 = *(const v4f*)(bptr0 + kp * 4);
            v4f b1v = *(const v4f*)(bptr1 + kp * 4);
            v2f alo = {a[0], a[1]},    ahi = {a[2], a[3]};
            v2f l0  = {b0[0], b0[1]},  h0  = {b0[2], b0[3]};
            v2f l1  = {b1v[0], b1v[1]}, h1 = {b1v[2], b1v[3]};
            c0 = __builtin_amdgcn_wmma_f32_16x16x4_f32(false, alo, false, l0, (short)0, c0, false, false);
            c1 = __builtin_amdgcn_wmma_f32_16x16x4_f32(false, alo, false, l1, (short)0, c1, false, false);
            c0 = __builtin_amdgcn_wmma_f32_16x16x4_f32(false, ahi, false, h0, (short)0, c0, false, false);
            c1 = __builtin_amdgcn_wmma_f32_16x16x4_f32(false, ahi, false, h1, (short)0, c1, false, false);
        }
        v8f cacc[2] = {c0, c1};

        // x_t for the 8 batch rows this lane owns (after matmul: lower peak pressure)
        float xv0[8], xv1[8], xv2[8];
        #pragma unroll
        for (int i = 0; i < 8; ++i) {
            int M = i + 8 * h;
            xv0[i] = xbuf[(rb * 16 + M) * 4 + 0];
            xv1[i] = xbuf[(rb * 16 + M) * 4 + 1];
            xv2[i] = xbuf[(rb * 16 + M) * 4 + 2];
        }

        // ---- LIF update on C-fragment layout (lane elem i <-> M = i + 8h, N = ncol[s]) ----
        #pragma unroll
        for (int s = 0; s < 2; ++s) {
            #pragma unroll
            for (int i = 0; i < 8; ++i) {
                int   M    = i + 8 * h;
                float cur1 = w10[s] * xv0[i] + w11[s] * xv1[i] + w12[s] * xv2[i];
                float base = BETA * mem[s][i] + cur1 + cacc[s][i] + bsum[s];
                float mnew = (mem[s][i] > THRESH) ? 0.0f : base;   // zero reset (old mem)
                mem[s][i]  = mnew;
                float spk  = (mnew > THRESH) ? 1.0f : 0.0f;
                if (nv[s]) {   // stores only — no WMMA under divergence
                    int n = ncol[s];
                    out_spk[(((size_t)t * BATCH) + bbase + M) * NHID + n] = spk;
                    // scatter into next step's A-fragment: k=n -> kc=n>>2, h2=(n>>1)&1, j=n&1
                    int lane2 = ((n & 2) << 3) + M;
                    spkf[((wb * 32) + lane2) * LSTRIDE + ((n >> 2) * 2) + (n & 1)] = spk;
                    if (spk != 0.0f) atomicAdd(acc + rb * 16 + M, w2n[s]);
                }
            }
        }
        __syncthreads();
    }

    // final cur2 readout for t = T-1
    if (wave == 0 && lane < 16) {
        float v = acc[((T_STEPS - 1) & 1) * 16 + lane];
        out_cur2[(size_t)(T_STEPS - 1) * BATCH + bbase + lane] = v + b2v;
    }
}

extern "C" void kernel_launch(void* const* d_in, const int* in_sizes, int n_in,
                              void* d_out, int out_size, void* d_ws, size_t ws_size,
                              hipStream_t stream) {
    (void)in_sizes; (void)n_in; (void)out_size; (void)d_ws; (void)ws_size;
    const float* x    = (const float*)d_in[0];
    const float* W1   = (const float*)d_in[1];
    const float* b1   = (const float*)d_in[2];
    const float* Wrec = (const float*)d_in[3];
    const float* brec = (const float*)d_in[4];
    const float* W2   = (const float*)d_in[5];
    const float* b2   = (const float*)d_in[6];
    float* out = (float*)d_out;

    static bool attr_set = false;  // idempotent host-side attribute, not a stream op
    if (!attr_set) {
        hipFuncSetAttribute((const void*)rsnn_persistent_kernel,
                            hipFuncAttributeMaxDynamicSharedMemorySize, LDS_BYTES);
        attr_set = true;
    }
    rsnn_persistent_kernel<<<dim3(BATCH / 16), dim3(BLOCK), LDS_BYTES, stream>>>(
        x, W1, b1, Wrec, brec, W2, b2, out);
}